// Dual_Quantize6_43645457662419
// MI455X (gfx1250) — compile-verified
//
#include <hip/hip_runtime.h>
#include <hip/hip_bf16.h>

typedef float v2f __attribute__((ext_vector_type(2)));
typedef float v8f __attribute__((ext_vector_type(8)));

#define DIM 256
#define NEMBED 1024
#define NROWS 65536                        // 16*64*64
#define QN ((size_t)NROWS * DIM)           // 16777216
#define DISTN ((size_t)NROWS * NEMBED)     // 67108864

// d_out layout (float elements), tuple order:
// q_hc_hr,q_hc_lr,q_lc_hr,q_lc_lr | 4 diffs | 4 ind | 4 dist
#define OUT_DIFF_BASE (4 * QN)
#define OUT_IND_BASE  (OUT_DIFF_BASE + 4)
#define OUT_DIST_BASE (OUT_IND_BASE + (size_t)4 * NROWS)

// ws layout (float elements)
#define WS_X2HR 0
#define WS_X2LR (WS_X2HR + NROWS)
#define WS_E2LR (WS_X2LR + NROWS)
#define WS_E2HR (WS_E2LR + NEMBED)
#define WS_ETLR (WS_E2HR + NEMBED)
#define WS_ETHR (WS_ETLR + DIM * NEMBED)
#define WS_PART (WS_ETHR + DIM * NEMBED)   // 4 * 4096 block partials

// ---------------- prep kernels ----------------

__global__ void row_sq_kernel(const float* __restrict__ xhr,
                              const float* __restrict__ xlr,
                              float* __restrict__ ws) {
  const float* X = (blockIdx.y == 0) ? xhr : xlr;
  float* out = ws + ((blockIdx.y == 0) ? WS_X2HR : WS_X2LR);
  const int wave = threadIdx.x >> 5, lane = threadIdx.x & 31;
  const int m = blockIdx.x * 8 + wave;
  const float* row = X + (size_t)m * DIM;
  float s = 0.f;
#pragma unroll
  for (int i = 0; i < DIM; i += 32) { const float v = row[i + lane]; s += v * v; }
#pragma unroll
  for (int off = 16; off > 0; off >>= 1) s += __shfl_xor(s, off, 32);
  if (lane == 0) out[m] = s;
}

__global__ void col_sq_kernel(const float* __restrict__ elr,
                              const float* __restrict__ ehr,
                              float* __restrict__ ws) {
  const float* E = (blockIdx.y == 0) ? elr : ehr;
  float* out = ws + ((blockIdx.y == 0) ? WS_E2LR : WS_E2HR);
  const int n = blockIdx.x * 256 + threadIdx.x;
  float s = 0.f;
  for (int d = 0; d < DIM; ++d) { const float v = E[(size_t)d * NEMBED + n]; s += v * v; }
  out[n] = s;
}

__global__ void transpose_e_kernel(const float* __restrict__ elr,
                                   const float* __restrict__ ehr,
                                   float* __restrict__ ws) {
  const float* E = (blockIdx.y == 0) ? elr : ehr;
  float* Et = ws + ((blockIdx.y == 0) ? WS_ETLR : WS_ETHR);
  const int n = blockIdx.x;
  const int d = threadIdx.x;
  Et[(size_t)n * DIM + d] = E[(size_t)d * NEMBED + n];
}

// ---------------- main GEMM + dist + argmin ----------------

#define BM 128
#define BN 128
#define KCH 64       // K chunk staged in LDS per barrier round
#define BFPAD 16     // pad so pair-row stride == 288 floats == 32 (mod 64) banks

__global__ __launch_bounds__(256) void gemm_dist_argmin_kernel(
    const float* __restrict__ xhr, const float* __restrict__ xlr,
    const float* __restrict__ elr, const float* __restrict__ ehr,
    const float* __restrict__ ws, float* __restrict__ out) {
  const int branch = blockIdx.y;  // 0:hc_hr 1:hc_lr 2:lc_hr 3:lc_lr
  const float* X  = (branch == 0 || branch == 2) ? xhr : xlr;
  const float* E  = (branch <= 1) ? ehr : elr;
  const float* x2 = ws + ((branch == 0 || branch == 2) ? WS_X2HR : WS_X2LR);
  const float* e2 = ws + ((branch <= 1) ? WS_E2HR : WS_E2LR);
  float* dist = out + OUT_DIST_BASE + (size_t)branch * DISTN;
  float* indf = out + OUT_IND_BASE + (size_t)branch * NROWS;

  // Pair-interleaved B tile: Bsf[k/2][n][0..1] = { B[2p][n], B[2p+1][n] }
  // -> each WMMA B fragment is one contiguous, 8B-aligned ds_load_b64.
  __shared__ float Bsf[KCH / 2][BN + BFPAD][2];
  __shared__ float rv[8][16][16];
  __shared__ int   ri[8][16][16];

  const int tid   = threadIdx.x;
  const int wave  = tid >> 5;
  const int lane  = tid & 31;
  const int lrow  = lane & 15;   // M (A) / N (B,C) within tile
  const int khalf = lane >> 4;   // K-pair selector for A/B operands

  const int m0 = blockIdx.x * BM;
  const int mw = m0 + wave * 16;

  float minv[8];
  int   mini[8];
  float x2r[8];
#pragma unroll
  for (int r = 0; r < 8; ++r) {
    minv[r] = 3.4e38f;
    mini[r] = 0;
    x2r[r]  = x2[mw + khalf * 8 + r];
  }

  const v8f vzero = {0.f, 0.f, 0.f, 0.f, 0.f, 0.f, 0.f, 0.f};

  for (int phase = 0; phase < NEMBED / BN; ++phase) {
    const int n0 = phase * BN;
    v8f acc[8];
#pragma unroll
    for (int t = 0; t < 8; ++t) acc[t] = vzero;

    for (int k0 = 0; k0 < DIM; k0 += KCH) {
      __syncthreads();
      // stage B chunk KCH x BN into pair-interleaved LDS
#pragma unroll
      for (int i = tid; i < KCH * (BN / 4); i += 256) {
        const int rrow = i >> 5;   // 0..KCH-1   (BN/4 == 32)
        const int c4   = i & 31;
        const float4 v = *reinterpret_cast<const float4*>(
            E + (size_t)(k0 + rrow) * NEMBED + n0 + c4 * 4);
        const int p = rrow >> 1, c = rrow & 1, nb = c4 * 4;
        Bsf[p][nb + 0][c] = v.x;
        Bsf[p][nb + 1][c] = v.y;
        Bsf[p][nb + 2][c] = v.z;
        Bsf[p][nb + 3][c] = v.w;
      }
      __syncthreads();

#pragma unroll
      for (int kk = 0; kk < KCH; kk += 4) {
        v2f a;
        {
          const float2 av = *reinterpret_cast<const float2*>(
              X + (size_t)(mw + lrow) * DIM + k0 + kk + khalf * 2);
          a.x = av.x; a.y = av.y;
        }
        const int p = (kk >> 1) + khalf;
#pragma unroll
        for (int t = 0; t < 8; ++t) {
          const v2f b = *reinterpret_cast<const v2f*>(&Bsf[p][t * 16 + lrow][0]);
          acc[t] = __builtin_amdgcn_wmma_f32_16x16x4_f32(
              false, a, false, b, (short)0, acc[t], false, false);
        }
      }
    }

    // epilogue: dist = x2 + e2 - 2*dot, stream out, running per-lane argmin
#pragma unroll
    for (int t = 0; t < 8; ++t) {
      const int n = n0 + t * 16 + lrow;
      const float e2n = e2[n];
#pragma unroll
      for (int r = 0; r < 8; ++r) {
        const int gm = mw + khalf * 8 + r;
        const float d = x2r[r] + e2n - 2.0f * acc[t][r];
        dist[(size_t)gm * NEMBED + n] = d;
        if (d < minv[r]) { minv[r] = d; mini[r] = n; }  // ascending n => first occurrence
      }
    }
  }

  // cross-lane argmin reduction (16 lanes share each row)
#pragma unroll
  for (int r = 0; r < 8; ++r) {
    const int row = khalf * 8 + r;
    rv[wave][row][lrow] = minv[r];
    ri[wave][row][lrow] = mini[r];
  }
  __syncthreads();
  if (tid < 128) {
    const int w = tid >> 4, row = tid & 15;
    float bv = rv[w][row][0];
    int   bi = ri[w][row][0];
#pragma unroll
    for (int l = 1; l < 16; ++l) {
      const float v = rv[w][row][l];
      const int   i = ri[w][row][l];
      if (v < bv || (v == bv && i < bi)) { bv = v; bi = i; }
    }
    indf[m0 + w * 16 + row] = (float)bi;
  }
}

// ---------------- gather q + loss partials ----------------

__global__ __launch_bounds__(256) void gather_q_diff_kernel(
    const float* __restrict__ xhr, const float* __restrict__ xlr,
    const float* __restrict__ ws, float* __restrict__ out,
    float* __restrict__ partials) {
  const int branch = blockIdx.y;
  const float* X  = (branch == 0 || branch == 2) ? xhr : xlr;
  const float* Et = ws + ((branch <= 1) ? WS_ETHR : WS_ETLR);
  const float* indf = out + OUT_IND_BASE + (size_t)branch * NROWS;
  float* q = out + (size_t)branch * QN;
  const int tid = threadIdx.x;
  const int m0 = blockIdx.x * 16;
  float s = 0.f;
#pragma unroll
  for (int j = 0; j < 16; ++j) {
    const int m = m0 + j;
    const int idx = (int)indf[m];
    const float qv = Et[(size_t)idx * DIM + tid];
    const float xv = X[(size_t)m * DIM + tid];
    q[(size_t)m * DIM + tid] = qv;
    const float df = qv - xv;
    s += df * df;
  }
  __shared__ float red[256];
  red[tid] = s;
  __syncthreads();
  for (int st = 128; st > 0; st >>= 1) {
    if (tid < st) red[tid] += red[tid + st];
    __syncthreads();
  }
  if (tid == 0) partials[branch * 4096 + blockIdx.x] = red[0];
}

__global__ void finalize_diff_kernel(const float* __restrict__ partials,
                                     float* __restrict__ out) {
  const int b = blockIdx.x;
  const int tid = threadIdx.x;
  float s = 0.f;
  for (int i = tid; i < 4096; i += 256) s += partials[b * 4096 + i];
  __shared__ float red[256];
  red[tid] = s;
  __syncthreads();
  for (int st = 128; st > 0; st >>= 1) {
    if (tid < st) red[tid] += red[tid + st];
    __syncthreads();
  }
  // diff = 0.25*mean + mean = 1.25 * sum / (N*DIM)
  if (tid == 0) out[OUT_DIFF_BASE + b] = 1.25f * red[0] / (float)QN;
}

// ---------------- launch ----------------

extern "C" void kernel_launch(void* const* d_in, const int* in_sizes, int n_in,
                              void* d_out, int out_size, void* d_ws, size_t ws_size,
                              hipStream_t stream) {
  (void)in_sizes; (void)n_in; (void)out_size; (void)ws_size;
  const float* xhr = (const float*)d_in[0];
  const float* xlr = (const float*)d_in[1];
  const float* elr = (const float*)d_in[2];
  const float* ehr = (const float*)d_in[3];
  float* out = (float*)d_out;
  float* ws  = (float*)d_ws;

  row_sq_kernel<<<dim3(NROWS / 8, 2), 256, 0, stream>>>(xhr, xlr, ws);
  col_sq_kernel<<<dim3(NEMBED / 256, 2), 256, 0, stream>>>(elr, ehr, ws);
  transpose_e_kernel<<<dim3(NEMBED, 2), DIM, 0, stream>>>(elr, ehr, ws);
  gemm_dist_argmin_kernel<<<dim3(NROWS / BM, 4), 256, 0, stream>>>(
      xhr, xlr, elr, ehr, ws, out);
  gather_q_diff_kernel<<<dim3(NROWS / 16, 4), 256, 0, stream>>>(
      xhr, xlr, ws, out, ws + WS_PART);
  finalize_diff_kernel<<<dim3(4), 256, 0, stream>>>(ws + WS_PART, out);
}